// GRNNTransformGated_24438363914723
// MI455X (gfx1250) — compile-verified
//
#include <hip/hip_runtime.h>

// Problem constants (from the reference)
#define L_LVL 16
#define N_NODES 200000
#define M_MERGE 100000
#define F_IN 7
#define H_DIM 64
#define K3 192   // 3*H
#define K4 256   // 4*H

// LDS row strides (floats), padded off multiples of 64 banks, multiples of 2
#define SH  200  // hhu / rhhu rows (192 used)
#define SHH 72   // h_H rows (64 used)
#define SZ  264  // z rows (256 used)

typedef __attribute__((ext_vector_type(2))) float v2f;
typedef __attribute__((ext_vector_type(8))) float v8f;

// ---------------------------------------------------------------------------
// A-fragment for V_WMMA_F32_16X16X4_F32 (16x4 fp32):
//   lane L holds row m = L%16; VGPR0 = K at k0 + (L/16)*2, VGPR1 = +1.
// B-fragment (4x16 fp32), mirrored: lane L holds col n = L%16, same K pairing.
// Since out[m][n] = sum_k A[m][k] * W[n][k]  (GEMM vs W^T), B[k][n] = W[n][k].
// ---------------------------------------------------------------------------
__device__ __forceinline__ v2f load_a_lds(const float* lds, int stride, int k0, int lane) {
  int row  = lane & 15;
  int koff = k0 + ((lane >> 4) << 1);
  const float* p = lds + row * stride + koff;   // 8B-aligned (stride, koff even)
  v2f a; a.x = p[0]; a.y = p[1];
  return a;
}

__device__ __forceinline__ v2f load_b_w(const float* __restrict__ W, int ldw,
                                        int nbase, int k0, int lane) {
  int n    = nbase + (lane & 15);
  int koff = k0 + ((lane >> 4) << 1);
  const float* p = W + n * ldw + koff;
  v2f b; b.x = p[0]; b.y = p[1];
  return b;
}

__device__ __forceinline__ float sigmoid_f(float x) {
  return 1.0f / (1.0f + __expf(-x));
}

// ---------------------------------------------------------------------------
// Leaf embedding: u[n][h] = tanh(sum_f C[n][f] * Wu[h][f] + bu[h])
// rows < split -> ubuf ; rows >= split -> dst (tail of new embedding buffer)
// Grid: N*H threads. K=7 is tiny: pure VALU, broadcast-friendly loads.
// ---------------------------------------------------------------------------
__global__ __launch_bounds__(256) void embed_kernel(
    const float* __restrict__ C, const float* __restrict__ Wu,
    const float* __restrict__ bu, float* __restrict__ ubuf,
    float* __restrict__ dst, int split) {
  int gid = blockIdx.x * blockDim.x + threadIdx.x;
  int n = gid >> 6;
  int h = gid & 63;
  float s = bu[h];
#pragma unroll
  for (int f = 0; f < F_IN; ++f)
    s += C[n * F_IN + f] * Wu[h * F_IN + f];
  float v = tanhf(s);
  if (n < split) ubuf[n * H_DIM + h] = v;
  else           dst[n * H_DIM + h]  = v;
}

// ---------------------------------------------------------------------------
// One scan step for 16 merge rows per workgroup (4 waves x 32 lanes).
// Phases: gather hhu -> r GEMM -> (sigmoid, r*hhu) -> h_H GEMM -> tanh ->
//         z GEMM (K=256 over [h_H | hhu]) -> softmax gate -> blend -> store.
// All GEMMs via V_WMMA_F32_16X16X4_F32, K-loop step 4.
// ---------------------------------------------------------------------------
__global__ __launch_bounds__(128) void merge_kernel(
    const float* __restrict__ emb,      // prev embedding, N x 64
    const float* __restrict__ ubuf,     // u rows [0,M), M x 64
    const long long* __restrict__ ch,   // children, M x 2 (int64)
    const float* __restrict__ Wr, const float* __restrict__ br,
    const float* __restrict__ Wh, const float* __restrict__ bh,
    const float* __restrict__ Wz, const float* __restrict__ bz,
    float* __restrict__ dst) {          // new embedding, rows [0,M)
  __shared__ float hhu [16 * SH];
  __shared__ float rhhu[16 * SH];
  __shared__ float hH  [16 * SHH];
  __shared__ float zb  [16 * SZ];

  const int tid  = threadIdx.x;
  const int wave = tid >> 5;
  const int lane = tid & 31;
  const int m0   = blockIdx.x * 16;

  // Warm weight rows this wave will stream (emits global_prefetch_b8).
  __builtin_prefetch(Wr + (wave * 48 + lane) * K3, 0, 0);
  __builtin_prefetch(Wz + (wave * 64 + lane) * K4, 0, 0);

  // ---- Phase 0: cooperative gather of hhu = [h_L | h_R | u_in] ------------
  for (int idx = tid; idx < 16 * K3; idx += 128) {
    int row = idx / K3, col = idx - row * K3;
    int m = m0 + row;
    float v;
    if (col < 64) {
      long long cl = ch[2 * m];
      v = emb[cl * H_DIM + col];
    } else if (col < 128) {
      long long cr = ch[2 * m + 1];
      v = emb[cr * H_DIM + (col - 64)];
    } else {
      v = ubuf[m * H_DIM + (col - 128)];
    }
    hhu[row * SH + col] = v;
  }
  __syncthreads();

  // ---- Phase 1: r = sigmoid(hhu @ Wr^T + br); rhhu = r * hhu --------------
  // 12 output tiles of 16 cols; wave owns tiles [wave*3 .. wave*3+2].
  {
    v8f acc[3] = {};
    for (int k0 = 0; k0 < K3; k0 += 4) {
      v2f a = load_a_lds(hhu, SH, k0, lane);
#pragma unroll
      for (int t = 0; t < 3; ++t) {
        v2f b = load_b_w(Wr, K3, (wave * 3 + t) * 16, k0, lane);
        acc[t] = __builtin_amdgcn_wmma_f32_16x16x4_f32(
            false, a, false, b, (short)0, acc[t], false, false);
      }
    }
    // C/D layout: VGPR v holds row (v + 8*(lane/16)), col = nbase + lane%16
    int rbase = (lane >> 4) << 3;
#pragma unroll
    for (int t = 0; t < 3; ++t) {
      int col = (wave * 3 + t) * 16 + (lane & 15);
      float bias = br[col];
#pragma unroll
      for (int v = 0; v < 8; ++v) {
        int row = rbase + v;
        float rv = sigmoid_f(acc[t][v] + bias);
        rhhu[row * SH + col] = rv * hhu[row * SH + col];
      }
    }
  }
  __syncthreads();

  // ---- Phase 2: h_H = tanh(rhhu @ Wh^T + bh) ------------------------------
  // 4 output tiles (64 cols); wave owns tile `wave`.
  {
    v8f acc = {};
    for (int k0 = 0; k0 < K3; k0 += 4) {
      v2f a = load_a_lds(rhhu, SH, k0, lane);
      v2f b = load_b_w(Wh, K3, wave * 16, k0, lane);
      acc = __builtin_amdgcn_wmma_f32_16x16x4_f32(
          false, a, false, b, (short)0, acc, false, false);
    }
    int rbase = (lane >> 4) << 3;
    int col = wave * 16 + (lane & 15);
    float bias = bh[col];
#pragma unroll
    for (int v = 0; v < 8; ++v) {
      int row = rbase + v;
      hH[row * SHH + col] = tanhf(acc[v] + bias);
    }
  }
  __syncthreads();

  // ---- Phase 3: z = [h_H | hhu] @ Wz^T + bz (K = 256) ---------------------
  // 16 output tiles; wave owns tiles [wave*4 .. wave*4+3] (= softmax group).
  {
    v8f acc[4] = {};
    for (int k0 = 0; k0 < K4; k0 += 4) {
      v2f a = (k0 < H_DIM) ? load_a_lds(hH, SHH, k0, lane)
                           : load_a_lds(hhu, SH, k0 - H_DIM, lane);
#pragma unroll
      for (int t = 0; t < 4; ++t) {
        v2f b = load_b_w(Wz, K4, wave * 64 + t * 16, k0, lane);
        acc[t] = __builtin_amdgcn_wmma_f32_16x16x4_f32(
            false, a, false, b, (short)0, acc[t], false, false);
      }
    }
    int rbase = (lane >> 4) << 3;
#pragma unroll
    for (int t = 0; t < 4; ++t) {
      int col = wave * 64 + t * 16 + (lane & 15);
      float bias = bz[col];
#pragma unroll
      for (int v = 0; v < 8; ++v) {
        int row = rbase + v;
        zb[row * SZ + col] = acc[t][v] + bias;
      }
    }
  }
  __syncthreads();

  // ---- Phase 4: 4-way softmax gate + blend + store ------------------------
  for (int idx = tid; idx < 16 * H_DIM; idx += 128) {
    int row = idx >> 6, j = idx & 63;
    float z0 = zb[row * SZ + j];
    float z1 = zb[row * SZ + 64 + j];
    float z2 = zb[row * SZ + 128 + j];
    float z3 = zb[row * SZ + 192 + j];
    float mx = fmaxf(fmaxf(z0, z1), fmaxf(z2, z3));
    float e0 = __expf(z0 - mx), e1 = __expf(z1 - mx);
    float e2 = __expf(z2 - mx), e3 = __expf(z3 - mx);
    float inv = 1.0f / (e0 + e1 + e2 + e3);
    float h = (e0 * hH[row * SHH + j]       // g0 * h_H
             + e1 * hhu[row * SH + j]       // g1 * h_L
             + e2 * hhu[row * SH + 64 + j]  // g2 * h_R
             + e3 * hhu[row * SH + 128 + j] // g3 * u_in
              ) * inv;
    dst[(m0 + row) * H_DIM + j] = h;
  }
}

// ---------------------------------------------------------------------------
// Host: 1 embed (level 0) + 15 x (embed, merge), serialized on `stream`.
// Ping-pong: buf[0] = workspace, buf[1] = d_out; 15 steps (odd) end in d_out.
// Workspace: N*H (embedding) + M*H (u scratch) floats = ~77 MB.
// ---------------------------------------------------------------------------
extern "C" void kernel_launch(void* const* d_in, const int* in_sizes, int n_in,
                              void* d_out, int out_size, void* d_ws, size_t ws_size,
                              hipStream_t stream) {
  const float*     contents = (const float*)d_in[0];
  const long long* children = (const long long*)d_in[1];
  const float* Wu = (const float*)d_in[2];
  const float* bu = (const float*)d_in[3];
  const float* Wh = (const float*)d_in[4];
  const float* bh = (const float*)d_in[5];
  const float* Wz = (const float*)d_in[6];
  const float* bz = (const float*)d_in[7];
  const float* Wr = (const float*)d_in[8];
  const float* br = (const float*)d_in[9];

  float* out  = (float*)d_out;
  float* ws0  = (float*)d_ws;                       // N*H floats (embedding)
  float* ubuf = ws0 + (size_t)N_NODES * H_DIM;      // M*H floats (u scratch)
  float* buf[2] = { ws0, out };

  const int embed_blocks = (N_NODES * H_DIM) / 256; // 50000
  const int merge_blocks = M_MERGE / 16;            // 6250

  // Level 0: emb0 = tanh(contents[0] @ Wu^T + bu), entirely into buf[0].
  embed_kernel<<<embed_blocks, 256, 0, stream>>>(
      contents, Wu, bu, buf[0], buf[0], N_NODES);

  for (int l = 1; l < L_LVL; ++l) {
    const float* src = buf[(l - 1) & 1];
    float*       dst = buf[l & 1];
    // u for level l: rows < M into ubuf, rows >= M straight into dst tail.
    embed_kernel<<<embed_blocks, 256, 0, stream>>>(
        contents + (size_t)l * N_NODES * F_IN, Wu, bu, ubuf, dst, M_MERGE);
    // Gated merge writes rows [0, M) of dst.
    merge_kernel<<<merge_blocks, 128, 0, stream>>>(
        src, ubuf, children + (size_t)(l - 1) * M_MERGE * 2,
        Wr, br, Wh, bh, Wz, bz, dst);
  }
}